// LMHTNeuron_80058190397906
// MI455X (gfx1250) — compile-verified
//
#include <hip/hip_runtime.h>
#include <hip/hip_bf16.h>

// LMHT spiking-neuron quantizer, MI455X (gfx1250).
// Memory-bound streaming recurrence: stage x[t] tiles via CDNA5 async
// global->LDS B128 copies (ASYNCcnt), compute 4-step recurrence in registers,
// non-temporal B128 stores for the spike output.

typedef float vf4 __attribute__((ext_vector_type(4)));

#define TSTEPS 4
#define BLOCK 256

__global__ __launch_bounds__(BLOCK) void lmht_neuron_kernel(
    const float* __restrict__ x,        // (T, ND) flat
    const float* __restrict__ scale_p,  // scalar
    const float* __restrict__ zp_p,     // scalar
    float* __restrict__ out,            // (T, ND) flat
    unsigned nd4)                       // ND / 4  (float4 count per timestep)
{
    // [t][tid] layout: lane stride 16B -> ds_read_b128 conflict-free.
    __shared__ vf4 smem[TSTEPS][BLOCK];

    const unsigned tid = threadIdx.x;
    const unsigned i   = blockIdx.x * BLOCK + tid;
    if (i >= nd4) return;

    const unsigned nd_bytes = nd4 * 16u;   // bytes per timestep plane (<2^31)
    const unsigned goff     = i * 16u;

    // Issue 4 async global->LDS copies (one per timestep); ASYNCcnt tracks them.
    #pragma unroll
    for (int t = 0; t < TSTEPS; ++t) {
        unsigned lds_addr = (unsigned)(unsigned long long)(&smem[t][tid]); // flat addr low 32 = LDS offset
        unsigned off      = goff + (unsigned)t * nd_bytes;
        asm volatile("global_load_async_to_lds_b128 %0, %1, %2"
                     :: "v"(lds_addr), "v"(off), "s"(x)
                     : "memory");
    }

    // Uniform scalar prep overlaps the async copies.
    float s       = scale_p[0];
    float z       = zp_p[0];
    float scale_c = fminf(fmaxf(s, 1e-4f), 1e4f);          // clamp_ste fwd
    float rzp     = fminf(fmaxf(rintf(z), 0.0f), 15.0f);   // round(RNE)+clamp fwd
    float Iz      = (rzp * 0.25f) * scale_c;               // (rzp/T)*scale
    float inv     = 1.0f / scale_c;                        // hoist the division

    float v[4];
    float v0 = 0.5f * scale_c;
    v[0] = v[1] = v[2] = v[3] = v0;

    // Drain the async engine before touching LDS.
    asm volatile("s_wait_asynccnt 0" ::: "memory");

    #pragma unroll
    for (int t = 0; t < TSTEPS; ++t) {
        vf4 xt = smem[t][tid];
        vf4 res;
        #pragma unroll
        for (int c = 0; c < 4; ++c) {
            float vv = v[c] + xt[c] + Iz;
            float q  = floorf(vv * inv);
            q        = fminf(fmaxf(q, 0.0f), 4.0f);        // clip to L_LEVELS
            float o  = q * scale_c;                        // multilevel output
            v[c]     = vv - o;                             // membrane reset
            res[c]   = o - Iz;                             // emitted spike
        }
        __builtin_nontemporal_store(
            res, reinterpret_cast<vf4*>(out) + (size_t)t * nd4 + i);
    }
}

extern "C" void kernel_launch(void* const* d_in, const int* in_sizes, int n_in,
                              void* d_out, int out_size, void* d_ws, size_t ws_size,
                              hipStream_t stream) {
    const float* x     = (const float*)d_in[0];
    const float* scale = (const float*)d_in[1];
    const float* zp    = (const float*)d_in[2];
    float* out         = (float*)d_out;

    long long total = in_sizes[0];        // T * N * D
    long long nd    = total / TSTEPS;     // N * D
    unsigned  nd4   = (unsigned)(nd / 4); // float4 tiles per timestep

    unsigned grid = (nd4 + BLOCK - 1) / BLOCK;
    hipLaunchKernelGGL(lmht_neuron_kernel, dim3(grid), dim3(BLOCK), 0, stream,
                       x, scale, zp, out, nd4);
}